// RSV_86543591014479
// MI455X (gfx1250) — compile-verified
//
#include <hip/hip_runtime.h>

// mean(|pred - mask|) over 16x1x1024x1024 fp32 tensors.
// The reference's Gaussian-conv loop is dead code w.r.t. the returned loss,
// so this is a pure streaming reduction: 128 MiB read @ 23.3 TB/s ~ 5.8 us.
// Wave-level reduction is done with v_wmma_f32_16x16x4_f32 (B = ones).

typedef __attribute__((ext_vector_type(2))) float v2f;
typedef __attribute__((ext_vector_type(4))) float v4f;
typedef __attribute__((ext_vector_type(8))) float v8f;

#define N_ELEM   (16u * 1024u * 1024u)   // 16 * 1 * 1024 * 1024
#define TPB      256                     // 8 waves of 32 on CDNA5
#define NBLOCKS  4096

__global__ __launch_bounds__(TPB)
void absdiff_mean_kernel(const float* __restrict__ pred,
                         const float* __restrict__ mask,
                         float* __restrict__ out) {
    const unsigned nvec   = N_ELEM / 4u;                       // float4 count
    const unsigned tid    = blockIdx.x * TPB + threadIdx.x;
    const unsigned stride = gridDim.x * TPB;

    const v4f* __restrict__ p4 = (const v4f*)pred;
    const v4f* __restrict__ m4 = (const v4f*)mask;

    float acc = 0.0f;
    for (unsigned i = tid; i < nvec; i += stride) {
        // single-pass stream: non-temporal B128 loads, don't pollute caches
        v4f p = __builtin_nontemporal_load(p4 + i);
        v4f m = __builtin_nontemporal_load(m4 + i);
        acc += __builtin_fabsf(p.x - m.x);
        acc += __builtin_fabsf(p.y - m.y);
        acc += __builtin_fabsf(p.z - m.z);
        acc += __builtin_fabsf(p.w - m.w);
    }

    // ---- wave32 reduction via WMMA: D(16x16) = A(16x4) * B(4x16) + 0 ----
    // A layout (32-bit, 2 VGPRs/lane): lanes 0-15 hold K=0,1 of row=lane,
    // lanes 16-31 hold K=2,3 of row=lane-16. Put the partial sum in the
    // first slot, zero the second: A[m] = [s_m, 0, s_{m+16}, 0].
    // With B = all-ones, D[m][n] = s_m + s_{m+16} for every n.
    v2f a; a.x = acc;  a.y = 0.0f;
    v2f b; b.x = 1.0f; b.y = 1.0f;
    v8f c = {};
    v8f d = __builtin_amdgcn_wmma_f32_16x16x4_f32(
        /*neg_a=*/false, a, /*neg_b=*/false, b,
        /*c_mod=*/(short)0, c, /*reuse_a=*/false, /*reuse_b=*/false);

    // Each lane sums its 8 D VGPRs: lanes 0-15 all hold sum of rows 0-7,
    // lanes 16-31 all hold sum of rows 8-15 (rows are constant across N).
    float r = 0.0f;
#pragma unroll
    for (int j = 0; j < 8; ++j) r += d[j];
    // combine the two half-wave totals
    r += __shfl_xor(r, 16, 32);

    // ---- block reduction (8 waves) + one atomic per block ----
    __shared__ float smem[TPB / 32];
    const int lane = threadIdx.x & 31;
    const int wid  = threadIdx.x >> 5;
    if (lane == 0) smem[wid] = r;
    __syncthreads();
    if (threadIdx.x == 0) {
        float bs = 0.0f;
#pragma unroll
        for (int j = 0; j < TPB / 32; ++j) bs += smem[j];
        atomicAdd(out, bs * (1.0f / (float)N_ELEM));
    }
}

extern "C" void kernel_launch(void* const* d_in, const int* in_sizes, int n_in,
                              void* d_out, int out_size, void* d_ws, size_t ws_size,
                              hipStream_t stream) {
    (void)in_sizes; (void)n_in; (void)out_size; (void)d_ws; (void)ws_size;
    const float* pred = (const float*)d_in[0];
    const float* mask = (const float*)d_in[1];
    float* out = (float*)d_out;

    // d_out is poisoned by the harness; zero it (capture-safe) before accumulate
    hipMemsetAsync(out, 0, sizeof(float), stream);
    absdiff_mean_kernel<<<NBLOCKS, TPB, 0, stream>>>(pred, mask, out);
}